// BERTEmbedder_28802050687525
// MI455X (gfx1250) — compile-verified
//
#include <hip/hip_runtime.h>
#include <hip/hip_bf16.h>
#include <stdint.h>

// Problem constants (match reference: B=16, S=4096, D=768)
#define BB 16
#define SS 4096
#define DD 768
#define TILE 8          // rows of a segment staged in LDS per async round
#define P2_THREADS 192  // 192 threads * float4 = 768 columns

typedef float v4f __attribute__((ext_vector_type(4)));

// ---------------------------------------------------------------------------
// CDNA5 async global->LDS DMA helpers (ASYNCcnt-tracked path)
// ---------------------------------------------------------------------------
__device__ __forceinline__ void async_load16_to_lds(const float* g, float* lds_ptr) {
    // vdst = 32-bit LDS byte offset (low 32 bits of generic shared pointer),
    // vaddr = 64-bit global address. ISA: GLOBAL_LOAD_ASYNC_TO_LDS_B128.
    uint32_t laddr = (uint32_t)(uintptr_t)lds_ptr;
    asm volatile("global_load_async_to_lds_b128 %0, %1, off"
                 :
                 : "v"(laddr), "v"(g)
                 : "memory");
}

__device__ __forceinline__ void wait_async_zero() {
#if __has_builtin(__builtin_amdgcn_s_wait_asynccnt)
    __builtin_amdgcn_s_wait_asynccnt(0);
#else
    asm volatile("s_wait_asynccnt 0" ::: "memory");
#endif
}

__device__ __forceinline__ void wait_ds_zero() {
    asm volatile("s_wait_dscnt 0" ::: "memory");
}

// ---------------------------------------------------------------------------
// Phase 1: per-batch token segmentation (block scan of 1-merge over valid prefix)
//   tok_start[b*(SS+1)+t]   = first subtoken index of token t
//   tok_start[b*(SS+1)+T]   = len (sentinel)
//   num_tokens[b]           = T
// ---------------------------------------------------------------------------
__global__ void seg_scan_kernel(const int* __restrict__ merge,
                                const int* __restrict__ lengths,
                                int* __restrict__ num_tokens,
                                int* __restrict__ tok_start) {
    const int b   = blockIdx.x;
    const int tid = threadIdx.x;        // 0..1023, each owns 4 positions

    int len = lengths[b];
    if (len < 1)  len = 1;
    if (len > SS) len = SS;

    const int s_base = tid * 4;
    int nt[4];
    int local = 0;
#pragma unroll
    for (int i = 0; i < 4; ++i) {
        const int s = s_base + i;
        int v = 0;
        if (s < len) {
            int m = (s == 0) ? 0 : (merge[b * SS + s] ? 1 : 0); // pos 0 starts a token
            v = 1 - m;                                          // 1 == new token
        }
        nt[i] = v;
        local += v;
    }

    __shared__ int sc[1024];
    sc[tid] = local;
    __syncthreads();

    // Hillis-Steele inclusive scan over per-thread sums
    for (int off = 1; off < 1024; off <<= 1) {
        int v  = sc[tid];
        int vp = (tid >= off) ? sc[tid - off] : 0;
        __syncthreads();
        sc[tid] = v + vp;
        __syncthreads();
    }

    const int excl = sc[tid] - local;   // exclusive prefix for this thread
    const int T    = sc[1023];          // total tokens in this batch row (>= 1)

    int run = excl;
#pragma unroll
    for (int i = 0; i < 4; ++i) {
        const int s = s_base + i;
        run += nt[i];
        if (nt[i]) tok_start[b * (SS + 1) + (run - 1)] = s;
    }

    if (tid == 0) {
        num_tokens[b] = T;
        tok_start[b * (SS + 1) + T] = len; // sentinel end for last token
    }
}

// ---------------------------------------------------------------------------
// Phase 2: one block per output token row. Stage segment rows into LDS via
// async DMA (each lane fetches the 16B it later consumes -> per-wave
// s_wait_asynccnt is the only sync needed), accumulate, scale by 1/count,
// write with non-temporal 128-bit stores. Rows t >= T write zeros.
// ---------------------------------------------------------------------------
__global__ __launch_bounds__(P2_THREADS)
void seg_mean_kernel(const float* __restrict__ hs,
                     const int* __restrict__ num_tokens,
                     const int* __restrict__ tok_start,
                     float* __restrict__ out) {
    const int b   = blockIdx.y;
    const int t   = blockIdx.x;
    const int tid = threadIdx.x;        // 0..191, column group = 4*tid

    __shared__ float stage[TILE][DD];

    v4f acc = {0.f, 0.f, 0.f, 0.f};

    const int T = num_tokens[b];
    if (t < T) {
        const int s0  = tok_start[b * (SS + 1) + t];
        const int s1  = tok_start[b * (SS + 1) + t + 1];
        const int cnt = s1 - s0;

        const float* base = hs + ((size_t)b * SS + (size_t)s0) * DD + 4 * tid;

        for (int r0 = 0; r0 < cnt; r0 += TILE) {
            const int n = (cnt - r0 < TILE) ? (cnt - r0) : TILE;
            // make sure prior-iteration LDS reads retired before DMA overwrites
            wait_ds_zero();
            for (int r = 0; r < n; ++r) {
                async_load16_to_lds(base + (size_t)(r0 + r) * DD, &stage[r][4 * tid]);
            }
            wait_async_zero();
            for (int r = 0; r < n; ++r) {
                acc += *(const v4f*)&stage[r][4 * tid];
            }
        }

        const float inv = 1.0f / (float)cnt;
        acc *= inv;
    }

    float* dst = out + ((size_t)b * SS + (size_t)t) * DD + 4 * tid;
    __builtin_nontemporal_store(acc, (v4f*)dst);
}

// ---------------------------------------------------------------------------
// Launch: inputs are [hidden_states f32, merge i32, lengths i32]
// ---------------------------------------------------------------------------
extern "C" void kernel_launch(void* const* d_in, const int* in_sizes, int n_in,
                              void* d_out, int out_size, void* d_ws, size_t ws_size,
                              hipStream_t stream) {
    (void)in_sizes; (void)n_in; (void)out_size; (void)ws_size;

    const float* hs      = (const float*)d_in[0];
    const int*   merge   = (const int*)d_in[1];
    const int*   lengths = (const int*)d_in[2];
    float*       out     = (float*)d_out;

    // Workspace layout: [num_tokens: BB ints][tok_start: BB*(SS+1) ints] (~262 KB)
    int* num_tokens = (int*)d_ws;
    int* tok_start  = num_tokens + BB;

    seg_scan_kernel<<<BB, 1024, 0, stream>>>(merge, lengths, num_tokens, tok_start);
    seg_mean_kernel<<<dim3(SS, BB), P2_THREADS, 0, stream>>>(hs, num_tokens, tok_start, out);
}